// GCN_36481452212961
// MI455X (gfx1250) — compile-verified
//
#include <hip/hip_runtime.h>
#include <math.h>

// GCN forward for MI455X (gfx1250, wave32).
// x:[N,64] f32, edge_index:[2,E] i32, W0/W1/W2:[64,64], b*:[64], Wc:[64,40], bc:[40]
// out: log_softmax(h @ Wc + bc) -> [N,40] f32

typedef __attribute__((ext_vector_type(2))) float v2f;
typedef __attribute__((ext_vector_type(8))) float v8f;

#define DIM 64
#define NCLS 40

// ---------------- degree / normalization ----------------
__global__ void gcn_deg_init(float* __restrict__ deg, int n) {
    int i = blockIdx.x * blockDim.x + threadIdx.x;
    if (i < n) deg[i] = 1.0f;                      // self-loop
}

__global__ void gcn_deg_scatter(const int* __restrict__ dst, float* __restrict__ deg, int nE) {
    int e = blockIdx.x * blockDim.x + threadIdx.x;
    if (e < nE) unsafeAtomicAdd(&deg[dst[e]], 1.0f);
}

__global__ void gcn_dinv(const float* __restrict__ deg, float* __restrict__ dinv, int n) {
    int i = blockIdx.x * blockDim.x + threadIdx.x;
    if (i < n) dinv[i] = rsqrtf(deg[i]);           // deg >= 1 always
}

// ---------------- dense transform: T = H @ W (fp32 WMMA), ACC = T * dinv^2 ----------------
// One wave computes a 16-row x 64-col strip. K looped in chunks of 4 using
// V_WMMA_F32_16X16X4_F32. W (64x64, 16 KB) staged in LDS.
__global__ __launch_bounds__(256) void gcn_gemm64_wmma(
    const float* __restrict__ Hin, const float* __restrict__ W,
    const float* __restrict__ dinv, float* __restrict__ T,
    float* __restrict__ ACC, int nRows)
{
    __shared__ float sW[DIM * DIM];
    {   // cooperative load of W: 1024 float4 / 256 threads
        const float4* Wv = (const float4*)W;
        float4* sWv = (float4*)sW;
        #pragma unroll
        for (int i = 0; i < 4; ++i) sWv[threadIdx.x + 256 * i] = Wv[threadIdx.x + 256 * i];
    }
    __syncthreads();

    const int wave = threadIdx.x >> 5;
    const int lane = threadIdx.x & 31;
    const int tile = blockIdx.x * 8 + wave;        // 16-row tile index
    if (tile * 16 >= nRows) return;

    const int row0 = tile * 16;
    const int l15  = lane & 15;
    const int hi   = lane >> 4;                    // lane half: 0 or 1
    const int kOff = hi * 2;                       // A frag: lanes 16-31 hold K+2,K+3
    const float* aBase = Hin + (size_t)(row0 + l15) * DIM + kOff;

    v8f acc0 = {}, acc1 = {}, acc2 = {}, acc3 = {};
    #pragma unroll
    for (int k0 = 0; k0 < DIM; k0 += 4) {
        v2f a = *(const v2f*)(aBase + k0);         // A[M=l15, K=k0+kOff .. +1]
        const float* bRow0 = &sW[(k0 + kOff) * DIM];
        const float* bRow1 = &sW[(k0 + kOff + 1) * DIM];
        v2f b;
        b.x = bRow0[l15 +  0]; b.y = bRow1[l15 +  0];
        acc0 = __builtin_amdgcn_wmma_f32_16x16x4_f32(false, a, false, b, (short)0, acc0, false, false);
        b.x = bRow0[l15 + 16]; b.y = bRow1[l15 + 16];
        acc1 = __builtin_amdgcn_wmma_f32_16x16x4_f32(false, a, false, b, (short)0, acc1, false, false);
        b.x = bRow0[l15 + 32]; b.y = bRow1[l15 + 32];
        acc2 = __builtin_amdgcn_wmma_f32_16x16x4_f32(false, a, false, b, (short)0, acc2, false, false);
        b.x = bRow0[l15 + 48]; b.y = bRow1[l15 + 48];
        acc3 = __builtin_amdgcn_wmma_f32_16x16x4_f32(false, a, false, b, (short)0, acc3, false, false);
    }

    // C/D layout: VGPR v -> row v (lanes 0-15) / row v+8 (lanes 16-31), col = lane&15.
    const int rBase = row0 + hi * 8;
    #pragma unroll
    for (int v = 0; v < 8; ++v) {
        const int r = rBase + v;
        const float d  = dinv[r];
        const float dd = d * d;                    // self-loop norm dinv[i]*dinv[i]
        const size_t o = (size_t)r * DIM + l15;
        float t0 = acc0[v], t1 = acc1[v], t2 = acc2[v], t3 = acc3[v];
        T[o +  0] = t0;  ACC[o +  0] = t0 * dd;
        T[o + 16] = t1;  ACC[o + 16] = t1 * dd;
        T[o + 32] = t2;  ACC[o + 32] = t2 * dd;
        T[o + 48] = t3;  ACC[o + 48] = t3 * dd;
    }
}

// ---------------- edge scatter: ACC[dst] += T[src] * dinv[src]*dinv[dst] ----------------
// 64 consecutive threads per edge -> contiguous 256B atomic burst per row.
__global__ void gcn_edge_scatter(const float* __restrict__ T,
                                 const float* __restrict__ dinv,
                                 const int* __restrict__ src,
                                 const int* __restrict__ dst,
                                 float* __restrict__ ACC, int nE)
{
    long long idx = (long long)blockIdx.x * blockDim.x + threadIdx.x;
    int e = (int)(idx >> 6);
    if (e >= nE) return;
    int c = (int)(idx & 63);
    int s = src[e], d = dst[e];
    float nrm = dinv[s] * dinv[d];
    unsafeAtomicAdd(&ACC[(size_t)d * DIM + c], T[(size_t)s * DIM + c] * nrm);
}

// ---------------- bias + ReLU ----------------
__global__ void gcn_bias_relu(const float* __restrict__ ACC, const float* __restrict__ b,
                              float* __restrict__ Hout, int total)
{
    int i = blockIdx.x * blockDim.x + threadIdx.x;
    if (i >= total) return;
    float v = ACC[i] + b[i & (DIM - 1)];
    Hout[i] = v > 0.f ? v : 0.f;
}

// ---------------- classifier + log_softmax ----------------
__global__ __launch_bounds__(256) void gcn_classifier(
    const float* __restrict__ H, const float* __restrict__ Wc,
    const float* __restrict__ bc, float* __restrict__ out, int n)
{
    __shared__ float sW[DIM * NCLS];
    __shared__ float sb[NCLS];
    for (int i = threadIdx.x; i < DIM * NCLS; i += 256) sW[i] = Wc[i];
    if (threadIdx.x < NCLS) sb[threadIdx.x] = bc[threadIdx.x];
    __syncthreads();

    int node = blockIdx.x * 256 + threadIdx.x;
    if (node >= n) return;

    float h[DIM];
    const float4* hp = (const float4*)(H + (size_t)node * DIM);
    #pragma unroll
    for (int k = 0; k < DIM / 4; ++k) ((float4*)h)[k] = hp[k];

    float lg[NCLS];
    float m = -3.402823466e38f;
    #pragma unroll 4
    for (int c = 0; c < NCLS; ++c) {
        float s = sb[c];
        #pragma unroll
        for (int k = 0; k < DIM; ++k) s = fmaf(h[k], sW[k * NCLS + c], s);
        lg[c] = s;
        m = fmaxf(m, s);
    }
    float sum = 0.f;
    #pragma unroll 4
    for (int c = 0; c < NCLS; ++c) sum += __expf(lg[c] - m);
    float lse = __logf(sum);
    float* o = out + (size_t)node * NCLS;
    #pragma unroll 4
    for (int c = 0; c < NCLS; ++c) o[c] = lg[c] - m - lse;
}

// ---------------- host orchestration ----------------
extern "C" void kernel_launch(void* const* d_in, const int* in_sizes, int n_in,
                              void* d_out, int out_size, void* d_ws, size_t ws_size,
                              hipStream_t stream)
{
    const float* x  = (const float*)d_in[0];
    const int*   ei = (const int*)  d_in[1];
    const float* W0 = (const float*)d_in[2];
    const float* b0 = (const float*)d_in[3];
    const float* W1 = (const float*)d_in[4];
    const float* b1 = (const float*)d_in[5];
    const float* W2 = (const float*)d_in[6];
    const float* b2 = (const float*)d_in[7];
    const float* Wc = (const float*)d_in[8];
    const float* bc = (const float*)d_in[9];
    float* out = (float*)d_out;

    const int N = in_sizes[0] / DIM;   // 100000
    const int E = in_sizes[1] / 2;     // 800000
    const int* srcIdx = ei;            // edge_index[0]
    const int* dstIdx = ei + E;        // edge_index[1]

    float* ws   = (float*)d_ws;
    float* deg  = ws;                          // N
    float* dinv = ws + (size_t)N;              // N
    float* T    = ws + (size_t)2 * N;          // N*64
    float* ACC  = T   + (size_t)N * DIM;       // N*64
    float* Hbuf = ACC + (size_t)N * DIM;       // N*64

    // 1) symmetric normalization factors
    gcn_deg_init   <<<(N + 255) / 256, 256, 0, stream>>>(deg, N);
    gcn_deg_scatter<<<(E + 255) / 256, 256, 0, stream>>>(dstIdx, deg, E);
    gcn_dinv       <<<(N + 255) / 256, 256, 0, stream>>>(deg, dinv, N);

    const int tiles      = (N + 15) / 16;
    const int gemmBlocks = (tiles + 7) / 8;
    const long long sThreads = (long long)E * DIM;
    const int sBlocks = (int)((sThreads + 255) / 256);
    const int total   = N * DIM;

    const float* Ws[3] = {W0, W1, W2};
    const float* bs[3] = {b0, b1, b2};
    const float* cur = x;
    for (int L = 0; L < 3; ++L) {
        gcn_gemm64_wmma <<<gemmBlocks, 256, 0, stream>>>(cur, Ws[L], dinv, T, ACC, N);
        gcn_edge_scatter<<<sBlocks, 256, 0, stream>>>(T, dinv, srcIdx, dstIdx, ACC, E);
        gcn_bias_relu   <<<(total + 255) / 256, 256, 0, stream>>>(ACC, bs[L], Hbuf, total);
        cur = Hbuf;
    }

    // 2) classifier + log_softmax
    gcn_classifier<<<(N + 255) / 256, 256, 0, stream>>>(Hbuf, Wc, bc, out, N);
}